// GDTLayer_15393162789294
// MI455X (gfx1250) — compile-verified
//
#include <hip/hip_runtime.h>

// ---------------------------------------------------------------------------
// GDT layer for gfx1250 (MI455X).  Sizes from the reference:
//   N = 50000 nodes (divisible by 16), D = 128, H = 8 heads x 16, FF = 512,
//   E = 850000 edges (self-loops guarantee in-degree >= 1), 5 PPR hops.
// fp32 throughout; GEMMs use V_WMMA_F32_16X16X4_F32.
// ---------------------------------------------------------------------------

#define D128   128
#define HEADS  8
#define HD     16
#define DFF    512
#define ALPHA  0.15f
#define SLOPE  0.2f

typedef __attribute__((ext_vector_type(2))) float v2f;
typedef __attribute__((ext_vector_type(8))) float v8f;

__device__ __forceinline__ v8f wmma4(v8f acc, v2f a, v2f b) {
  // D = A(16x4) * B(4x16) + C, fp32.  8-arg form: neg_a, A, neg_b, B, c_mod, C, reuse_a, reuse_b
  return __builtin_amdgcn_wmma_f32_16x16x4_f32(false, a, false, b, (short)0, acc,
                                               false, false);
}

// monotonic float<->uint encoding so unsigned atomicMax == float max
__device__ __forceinline__ unsigned enc_f(float f) {
  unsigned b = __float_as_uint(f);
  return (b & 0x80000000u) ? ~b : (b | 0x80000000u);
}
__device__ __forceinline__ float dec_f(unsigned e) {
  unsigned b = (e & 0x80000000u) ? (e ^ 0x80000000u) : ~e;
  return __uint_as_float(b);
}

// ---------------------------------------------------------------------------
// LayerNorm over 128-wide rows: one wave per row, float4 per lane.
// ---------------------------------------------------------------------------
__global__ void k_layernorm(const float* __restrict__ in, const float* __restrict__ g,
                            const float* __restrict__ b, float* __restrict__ out,
                            int nrows) {
  int row  = blockIdx.x * 8 + (threadIdx.x >> 5);
  int lane = threadIdx.x & 31;
  if (row >= nrows) return;
  float4 v = ((const float4*)(in + (size_t)row * D128))[lane];
  float s  = v.x + v.y + v.z + v.w;
  float sq = v.x * v.x + v.y * v.y + v.z * v.z + v.w * v.w;
  for (int off = 16; off > 0; off >>= 1) {
    s  += __shfl_xor(s,  off, 32);
    sq += __shfl_xor(sq, off, 32);
  }
  float mu  = s * (1.0f / 128.0f);
  float var = sq * (1.0f / 128.0f) - mu * mu;
  float inv = rsqrtf(var + 1e-5f);
  float4 gg = ((const float4*)g)[lane];
  float4 bb = ((const float4*)b)[lane];
  float4 o;
  o.x = (v.x - mu) * inv * gg.x + bb.x;
  o.y = (v.y - mu) * inv * gg.y + bb.y;
  o.z = (v.z - mu) * inv * gg.z + bb.z;
  o.w = (v.w - mu) * inv * gg.w + bb.w;
  ((float4*)(out + (size_t)row * D128))[lane] = o;
}

// ---------------------------------------------------------------------------
// C[M x 128] = A[M x 128] @ W^T   (W row-major [128 x 128])
// Block = 256 thr (8 waves); block owns 16 rows; wave w owns cols [16w,16w+16).
// M % 16 == 0 (N = 50000 = 3125*16).
// ---------------------------------------------------------------------------
__global__ void k_gemm128(const float* __restrict__ A, const float* __restrict__ W,
                          float* __restrict__ C, int M) {
  __shared__ float sA[16 * D128];
  int rowblk = blockIdx.x * 16;
  for (int i = threadIdx.x; i < 16 * D128 / 4; i += 256)
    ((float4*)sA)[i] = ((const float4*)(A + (size_t)rowblk * D128))[i];
  __syncthreads();

  int wave = threadIdx.x >> 5;
  int lane = threadIdx.x & 31;
  int l16  = lane & 15;
  int half = lane >> 4;
  int kph  = half << 1;                 // lanes 16-31 carry K+2/K+3
  int col0 = wave * 16;

  v8f acc = {};
  for (int k0 = 0; k0 < D128; k0 += 4) {
    int k = k0 + kph;
    v2f a, b;
    a.x = sA[l16 * D128 + k];
    a.y = sA[l16 * D128 + k + 1];
    b.x = W[(size_t)(col0 + l16) * D128 + k];
    b.y = W[(size_t)(col0 + l16) * D128 + k + 1];
    acc = wmma4(acc, a, b);
  }
#pragma unroll
  for (int r = 0; r < 8; ++r)
    C[(size_t)(rowblk + r + 8 * half) * D128 + col0 + l16] = acc[r];
}

// ---------------------------------------------------------------------------
// Edge pipeline scratch init: deg = 0, emax = enc(-inf sentinel) = 0, esum = 0
// ---------------------------------------------------------------------------
__global__ void k_init(int* __restrict__ deg, unsigned* __restrict__ emax,
                       float* __restrict__ esum, int n) {
  int i = blockIdx.x * blockDim.x + threadIdx.x;
  if (i < n * HEADS) { emax[i] = 0u; esum[i] = 0.0f; }
  if (i < n) deg[i] = 0;
}

__global__ void k_degree(const int* __restrict__ dst, int* __restrict__ deg, int E) {
  int e = blockIdx.x * blockDim.x + threadIdx.x;
  if (e < E) atomicAdd(&deg[dst[e]], 1);
}

// ---------------------------------------------------------------------------
// Attention logits: one thread per (edge, head).
//   e = sum_d lrelu(fh[src]*ft[dst]) * attn  ;  * log(deg[dst]) / HD
// also segment-max over dst via monotonic-encoded unsigned atomicMax.
// ---------------------------------------------------------------------------
__global__ void k_logits(const float* __restrict__ fh, const float* __restrict__ ft,
                         const int* __restrict__ src, const int* __restrict__ dst,
                         const float* __restrict__ attn, const int* __restrict__ deg,
                         float* __restrict__ elog, unsigned* __restrict__ emax, int E) {
  int tid = blockIdx.x * blockDim.x + threadIdx.x;
  if (tid >= E * HEADS) return;
  int e = tid >> 3, h = tid & 7;
  int s = src[e], d = dst[e];
  const float* u  = fh   + (size_t)s * D128 + h * HD;
  const float* v  = ft   + (size_t)d * D128 + h * HD;
  const float* aw = attn + h * HD;
  float acc = 0.0f;
#pragma unroll
  for (int i = 0; i < HD; ++i) {
    float p = u[i] * v[i];
    p = (p > 0.0f) ? p : SLOPE * p;
    acc += p * aw[i];
  }
  acc *= logf((float)deg[d]) * (1.0f / (float)HD);
  elog[tid] = acc;
  atomicMax(&emax[d * HEADS + h], enc_f(acc));
}

__global__ void k_expsum(float* __restrict__ elog, const int* __restrict__ dst,
                         const unsigned* __restrict__ emax, float* __restrict__ esum,
                         int E) {
  int tid = blockIdx.x * blockDim.x + threadIdx.x;
  if (tid >= E * HEADS) return;
  int e = tid >> 3, h = tid & 7;
  int d = dst[e];
  float m  = dec_f(emax[d * HEADS + h]);
  float ex = __expf(elog[tid] - m);
  elog[tid] = ex;
  atomicAdd(&esum[d * HEADS + h], ex);
}

// a = ex / esum, with (1-ALPHA) folded in so each hop is a single scatter pass
__global__ void k_norm(float* __restrict__ elog, const int* __restrict__ dst,
                       const float* __restrict__ esum, int E) {
  int tid = blockIdx.x * blockDim.x + threadIdx.x;
  if (tid >= E * HEADS) return;
  int e = tid >> 3, h = tid & 7;
  elog[tid] = elog[tid] / esum[dst[e] * HEADS + h] * (1.0f - ALPHA);
}

// fB = ALPHA * feat0   (per-hop re-init; scatter adds (1-ALPHA)*agg on top)
__global__ void k_ppr_init(float* __restrict__ fB, const float* __restrict__ fe, int n) {
  int i = blockIdx.x * blockDim.x + threadIdx.x;
  if (i < n) fB[i] = ALPHA * fe[i];
}

// One wave per edge: 32 lanes x float4 = the edge's full 128-float message.
// Gather f[src] (contiguous 512B), weight by a[e,head], atomic-add into f[dst]
// (contiguous 512B region -> coalesced L2 atomics; arrays are L2-resident).
__global__ void k_scatter(const float* __restrict__ fA, const float* __restrict__ a,
                          const int* __restrict__ src, const int* __restrict__ dst,
                          float* __restrict__ fB, int E) {
  int tid = blockIdx.x * blockDim.x + threadIdx.x;
  int e = tid >> 5, c = tid & 31;               // c: which float4 chunk (0..31)
  if (e >= E) return;
  int s = src[e], d = dst[e];
  float w = a[e * HEADS + (c >> 2)];            // head = chunk/4
  float4 vs = ((const float4*)(fA + (size_t)s * D128))[c];
  float* pd = fB + (size_t)d * D128 + c * 4;
  atomicAdd(pd + 0, w * vs.x);
  atomicAdd(pd + 1, w * vs.y);
  atomicAdd(pd + 2, w * vs.z);
  atomicAdd(pd + 3, w * vs.w);
}

__global__ void k_resid(const float* __restrict__ f, const float* __restrict__ feat,
                        float* __restrict__ rst, int n) {
  int i = blockIdx.x * blockDim.x + threadIdx.x;
  if (i < n) rst[i] = f[i] + feat[i];
}

// ---------------------------------------------------------------------------
// Fused FFN per 16-row block: mid = relu(x@W1^T + b1) lives entirely in LDS,
// out = mid@W2^T + b2 + resid.  W1:[512x128], W2:[128x512] row-major.
// ---------------------------------------------------------------------------
__global__ void k_ffn(const float* __restrict__ x, const float* __restrict__ W1,
                      const float* __restrict__ b1, const float* __restrict__ W2,
                      const float* __restrict__ b2, const float* __restrict__ resid,
                      float* __restrict__ out, int M) {
  __shared__ float sX[16 * D128];     //  8 KB
  __shared__ float sMid[16 * DFF];    // 32 KB
  int rowblk = blockIdx.x * 16;
  for (int i = threadIdx.x; i < 16 * D128 / 4; i += 256)
    ((float4*)sX)[i] = ((const float4*)(x + (size_t)rowblk * D128))[i];
  __syncthreads();

  int wave = threadIdx.x >> 5;
  int lane = threadIdx.x & 31;
  int l16  = lane & 15;
  int half = lane >> 4;
  int kph  = half << 1;

  // FF1: wave owns 4 column tiles -> cols [64*wave, 64*wave+64)
  for (int t = 0; t < 4; ++t) {
    int col0 = wave * 64 + t * 16;
    v8f acc = {};
    for (int k0 = 0; k0 < D128; k0 += 4) {
      int k = k0 + kph;
      v2f a, b;
      a.x = sX[l16 * D128 + k];
      a.y = sX[l16 * D128 + k + 1];
      b.x = W1[(size_t)(col0 + l16) * D128 + k];
      b.y = W1[(size_t)(col0 + l16) * D128 + k + 1];
      acc = wmma4(acc, a, b);
    }
    float bias = b1[col0 + l16];
#pragma unroll
    for (int r = 0; r < 8; ++r) {
      float v = acc[r] + bias;
      sMid[(r + 8 * half) * DFF + col0 + l16] = (v > 0.0f) ? v : 0.0f;
    }
  }
  __syncthreads();

  // FF2: wave owns one 16x16 output tile, K = 512 from LDS
  int col0 = wave * 16;
  v8f acc = {};
  for (int k0 = 0; k0 < DFF; k0 += 4) {
    int k = k0 + kph;
    v2f a, b;
    a.x = sMid[l16 * DFF + k];
    a.y = sMid[l16 * DFF + k + 1];
    b.x = W2[(size_t)(col0 + l16) * DFF + k];
    b.y = W2[(size_t)(col0 + l16) * DFF + k + 1];
    acc = wmma4(acc, a, b);
  }
  float bias = b2[col0 + l16];
#pragma unroll
  for (int r = 0; r < 8; ++r) {
    size_t idx = (size_t)(rowblk + r + 8 * half) * D128 + col0 + l16;
    out[idx] = acc[r] + bias + resid[idx];
  }
}

// ---------------------------------------------------------------------------
extern "C" void kernel_launch(void* const* d_in, const int* in_sizes, int n_in,
                              void* d_out, int out_size, void* d_ws, size_t ws_size,
                              hipStream_t stream) {
  const float* feat  = (const float*)d_in[0];
  const int*   src   = (const int*)  d_in[1];
  const int*   dst   = (const int*)  d_in[2];
  const float* ln1_g = (const float*)d_in[3];
  const float* ln1_b = (const float*)d_in[4];
  const float* Whd   = (const float*)d_in[5];
  const float* Wtl   = (const float*)d_in[6];
  const float* Wen   = (const float*)d_in[7];
  const float* attn  = (const float*)d_in[8];
  const float* ln2_g = (const float*)d_in[9];
  const float* ln2_b = (const float*)d_in[10];
  const float* Wff1  = (const float*)d_in[11];
  const float* bff1  = (const float*)d_in[12];
  const float* Wff2  = (const float*)d_in[13];
  const float* bff2  = (const float*)d_in[14];

  const int N = in_sizes[0] / D128;
  const int E = in_sizes[1];
  const size_t nf = (size_t)N * D128;

  char* w = (char*)d_ws;
  float* bufH  = (float*)w; w += nf * 4;                 // h, later rst
  float* bufA  = (float*)w; w += nf * 4;                 // fh, later hop ping
  float* bufB  = (float*)w; w += nf * 4;                 // ft, later hop pong
  float* bufFE = (float*)w; w += nf * 4;                 // fe == feat0
  float* bufX  = (float*)w; w += nf * 4;                 // LN2 output
  float* elog  = (float*)w; w += (size_t)E * HEADS * 4;  // logits -> ex -> a
  int*      deg  = (int*)w;      w += (size_t)N * 4;
  unsigned* emax = (unsigned*)w; w += (size_t)N * HEADS * 4;
  float*    esum = (float*)w;    w += (size_t)N * HEADS * 4;

  const int TB = 256;
  const int rowBlocks = (N + 15) / 16;

  // 1) LN1
  k_layernorm<<<(N + 7) / 8, TB, 0, stream>>>(feat, ln1_g, ln1_b, bufH, N);
  // 2) projections (WMMA fp32)
  k_gemm128<<<rowBlocks, TB, 0, stream>>>(bufH, Whd, bufA, N);   // fh
  k_gemm128<<<rowBlocks, TB, 0, stream>>>(bufH, Wtl, bufB, N);   // ft
  k_gemm128<<<rowBlocks, TB, 0, stream>>>(bufH, Wen, bufFE, N);  // fe
  // 3) edge softmax
  k_init<<<(N * HEADS + TB - 1) / TB, TB, 0, stream>>>(deg, emax, esum, N);
  k_degree<<<(E + TB - 1) / TB, TB, 0, stream>>>(dst, deg, E);
  k_logits<<<(E * HEADS + TB - 1) / TB, TB, 0, stream>>>(bufA, bufB, src, dst, attn,
                                                         deg, elog, emax, E);
  k_expsum<<<(E * HEADS + TB - 1) / TB, TB, 0, stream>>>(elog, dst, emax, esum, E);
  k_norm<<<(E * HEADS + TB - 1) / TB, TB, 0, stream>>>(elog, dst, esum, E);
  // 4) 5 PPR hops (ping-pong; fh/ft buffers are free now)
  const float* fin = bufFE;
  float* fout = bufA;
  for (int hop = 0; hop < 5; ++hop) {
    k_ppr_init<<<((int)nf + TB - 1) / TB, TB, 0, stream>>>(fout, bufFE, (int)nf);
    k_scatter<<<((E * 32) + TB - 1) / TB, TB, 0, stream>>>(fin, elog, src, dst, fout, E);
    fin  = fout;
    fout = (fout == bufA) ? bufB : bufA;
  }
  // 5) residual, LN2, fused FFN (+ residual) -> d_out
  k_resid<<<((int)nf + TB - 1) / TB, TB, 0, stream>>>(fin, feat, bufH, (int)nf);
  k_layernorm<<<(N + 7) / 8, TB, 0, stream>>>(bufH, ln2_g, ln2_b, bufX, N);
  k_ffn<<<rowBlocks, TB, 0, stream>>>(bufX, Wff1, bff1, Wff2, bff2, bufH,
                                      (float*)d_out, N);
}